// Rnn_52338471469728
// MI455X (gfx1250) — compile-verified
//
#include <hip/hip_runtime.h>
#include <math.h>

#define Bsz 64
#define Ssz 256
#define Csz 512
#define Hsz 1024
#define Osz 1024
#define Lsz 4
#define NI  3   // L-1 inner recurrent layers

typedef __attribute__((ext_vector_type(2))) float v2f;
typedef __attribute__((ext_vector_type(8))) float v8f;

__device__ __forceinline__ v8f wmma_f32(v2f a, v2f b, v8f c) {
  // D = A(16x4) * B(4x16) + C   (full fp32 matrix op on CDNA5)
  return __builtin_amdgcn_wmma_f32_16x16x4_f32(false, a, false, b, (short)0, c,
                                               false, false);
}

// Low 32 bits of a generic pointer to __shared__ = wave-relative LDS byte addr.
__device__ __forceinline__ unsigned lds_addr(const volatile void* p) {
  return (unsigned)(uintptr_t)p;
}

// GLOBAL_LOAD_ASYNC_TO_LDS_B64: copies 8 bytes from global memory to LDS, tracked by ASYNCcnt.
__device__ __forceinline__ void async_copy_b64(unsigned lds, const float* g) {
  asm volatile("global_load_async_to_lds_b64 %0, %1, off"
               :: "v"(lds), "v"(g) : "memory");
}

__device__ __forceinline__ void wait_async0() {
  asm volatile("s_wait_asynccnt 0x0" ::: "memory");
}

// ---------------------------------------------------------------------------
// Stage one 32-wide K slab of A(64 rows) and W(64 rows) into LDS slabs
// (both row-major [row][k], pad 34). 128 threads, 16 b64 copies each.
// ---------------------------------------------------------------------------
__device__ __forceinline__ void stage64(
    const float* __restrict__ A, int lda,
    const float* __restrict__ W, int ldw, int k0,
    float (*As)[34], float (*Bs)[34])
{
  const int tid = threadIdx.x;
  #pragma unroll
  for (int j = 0; j < 16; ++j) {
    const int p = tid + j * 128;            // pair index; branch uniform per j
    if (p < 1024) {
      const int r = p >> 4, c = (p & 15) << 1;
      async_copy_b64(lds_addr(&As[r][c]), A + (size_t)r * lda + k0 + c);
    } else {
      const int q = p - 1024;
      const int r = q >> 4, c = (q & 15) << 1;
      async_copy_b64(lds_addr(&Bs[r][c]), W + (size_t)r * ldw + k0 + c);
    }
  }
}

// ---------------------------------------------------------------------------
// 64(M) x 64(N) tile GEMM with double-buffered async LDS staging:
//   D += A(64xK) * W(64xK)^T,  A[m][k] stride lda, W[n][k] stride ldw
// block = 128 threads (4 waves); wave w owns output cols [w*16, w*16+16)
// acc[m] = 16x16 fragment for rows [m*16, m*16+16)
// ---------------------------------------------------------------------------
__device__ __forceinline__ void gemm_tile64_db(
    const float* __restrict__ A, int lda,
    const float* __restrict__ W, int ldw, int K,
    float (*As)[64][34], float (*Bs)[64][34], v8f* acc)
{
  const int lane = threadIdx.x & 31;
  const int wave = threadIdx.x >> 5;
  const int l16 = lane & 15, hi = lane >> 4;
  const int nw  = wave * 16;

  __syncthreads();                       // previous user of the slabs is done
  stage64(A, lda, W, ldw, 0, As[0], Bs[0]);

  for (int k0 = 0; k0 < K; k0 += 32) {
    const int p = (k0 >> 5) & 1;
    wait_async0();                       // my async fills of slab p complete
    __syncthreads();                     // everyone's fills done; slab p^1 free
    if (k0 + 32 < K)
      stage64(A, lda, W, ldw, k0 + 32, As[p ^ 1], Bs[p ^ 1]);
    #pragma unroll
    for (int kk = 0; kk < 8; ++kk) {
      const int k = kk * 4 + hi * 2;     // lanes 0-15: k,k+1 ; 16-31: k+2,k+3
      v2f bf;
      bf.x = Bs[p][nw + l16][k];
      bf.y = Bs[p][nw + l16][k + 1];
      #pragma unroll
      for (int m = 0; m < 4; ++m) {
        v2f af;
        af.x = As[p][m * 16 + l16][k];
        af.y = As[p][m * 16 + l16][k + 1];
        acc[m] = wmma_f32(af, bf, acc[m]);
      }
    }
  }
}

// ---------------------------------------------------------------------------
// One-time: c1[b,h] = h0[b,:] @ fc1_w[h, C:C+H]^T + fc1_b[h]   (grid 16x128)
// ---------------------------------------------------------------------------
__global__ __launch_bounds__(128) void rnn_init_c1(
    const float* __restrict__ hid, const float* __restrict__ fc1_w,
    const float* __restrict__ fc1_b, float* __restrict__ c1)
{
  __shared__ float As[2][64][34];
  __shared__ float Bs[2][64][34];
  const int lane = threadIdx.x & 31;
  const int wave = threadIdx.x >> 5;
  const int l16 = lane & 15, hi = lane >> 4;
  const int n0  = blockIdx.x * 64;
  const int col = n0 + wave * 16 + l16;
  const v8f vz = {0.f, 0.f, 0.f, 0.f, 0.f, 0.f, 0.f, 0.f};
  v8f acc[4] = {vz, vz, vz, vz};

  // A = h0 : hid[b*L*H + 0*H + k]
  gemm_tile64_db(hid, Lsz * Hsz,
                 fc1_w + (size_t)n0 * (Hsz + Csz) + Csz, Hsz + Csz, Hsz,
                 As, Bs, acc);

  const float bias = fc1_b[col];
  #pragma unroll
  for (int m = 0; m < 4; ++m)
    #pragma unroll
    for (int r = 0; r < 8; ++r) {
      int row = m * 16 + hi * 8 + r;
      c1[row * Hsz + col] = acc[m][r] + bias;
    }
}

// One-time: carry init  hbuf0[l][b][h] = hid[b][l+1][h]
__global__ void rnn_init_h(const float* __restrict__ hid, float* __restrict__ h0buf)
{
  int idx = blockIdx.x * blockDim.x + threadIdx.x;
  if (idx >= NI * Bsz * Hsz) return;
  int l = idx / (Bsz * Hsz);
  int rem = idx % (Bsz * Hsz);
  int b = rem / Hsz, h = rem % Hsz;
  h0buf[idx] = hid[(size_t)b * Lsz * Hsz + (size_t)(l + 1) * Hsz + h];
}

// ---------------------------------------------------------------------------
// Step kernel 1: for a 64x64 (b,h) tile, compute
//   hn[l] = hr[l] @ lin_w[l]^T + lin_b[l]        (carry for next step)
//   temp  = tanh(hn[2] + tanh(hn[1] + tanh(hn[0] + tanh(x_t@W1x^T + c1))))
// grid 16 x 128
// ---------------------------------------------------------------------------
__global__ __launch_bounds__(128) void rnn_step1(
    const float* __restrict__ x, const float* __restrict__ c1,
    const float* __restrict__ hr, const float* __restrict__ fc1_w,
    const float* __restrict__ lin_w, const float* __restrict__ lin_b,
    float* __restrict__ hn, float* __restrict__ temp, int t)
{
  __shared__ float As[2][64][34];
  __shared__ float Bs[2][64][34];
  const int lane = threadIdx.x & 31;
  const int wave = threadIdx.x >> 5;
  const int l16 = lane & 15, hi = lane >> 4;
  const int n0  = blockIdx.x * 64;
  const int col = n0 + wave * 16 + l16;
  const v8f vz = {0.f, 0.f, 0.f, 0.f, 0.f, 0.f, 0.f, 0.f};

  float tv[4][8];
  {
    v8f acc[4] = {vz, vz, vz, vz};
    // x-part of fc1: A[b][k] = x[b*S*C + t*C + k]
    gemm_tile64_db(x + (size_t)t * Csz, Ssz * Csz,
                   fc1_w + (size_t)n0 * (Hsz + Csz), Hsz + Csz, Csz,
                   As, Bs, acc);
    #pragma unroll
    for (int m = 0; m < 4; ++m)
      #pragma unroll
      for (int r = 0; r < 8; ++r) {
        int row = m * 16 + hi * 8 + r;
        tv[m][r] = tanhf(acc[m][r] + c1[row * Hsz + col]);
      }
  }

  for (int l = 0; l < NI; ++l) {
    v8f acc[4] = {vz, vz, vz, vz};
    gemm_tile64_db(hr + (size_t)l * Bsz * Hsz, Hsz,
                   lin_w + (size_t)l * Hsz * Hsz + (size_t)n0 * Hsz, Hsz, Hsz,
                   As, Bs, acc);
    const float bb = lin_b[l * Hsz + col];
    #pragma unroll
    for (int m = 0; m < 4; ++m)
      #pragma unroll
      for (int r = 0; r < 8; ++r) {
        int row = m * 16 + hi * 8 + r;
        float hv = acc[m][r] + bb;
        hn[(size_t)l * Bsz * Hsz + row * Hsz + col] = hv;
        tv[m][r] = tanhf(hv + tv[m][r]);
      }
  }

  #pragma unroll
  for (int m = 0; m < 4; ++m)
    #pragma unroll
    for (int r = 0; r < 8; ++r) {
      int row = m * 16 + hi * 8 + r;
      temp[row * Hsz + col] = tv[m][r];
    }
}

// ---------------------------------------------------------------------------
// Step kernel 2 staging: A slab 16x32, B slab 128x32 (both row-major, pad 34).
// 256 threads, 9 b64 copies each.
// ---------------------------------------------------------------------------
__device__ __forceinline__ void stage2(
    const float* __restrict__ Arow, const float* __restrict__ Wc, int k0,
    float (*As)[34], float (*Bs)[34])
{
  const int tid = threadIdx.x;
  #pragma unroll
  for (int j = 0; j < 9; ++j) {
    const int p = tid + j * 256;            // branch uniform per j
    if (p < 256) {
      const int r = p >> 4, c = (p & 15) << 1;
      async_copy_b64(lds_addr(&As[r][c]), Arow + (size_t)r * Hsz + k0 + c);
    } else {
      const int q = p - 256;
      const int r = q >> 4, c = (q & 15) << 1;
      async_copy_b64(lds_addr(&Bs[r][c]), Wc + (size_t)r * Hsz + k0 + c);
    }
  }
}

// ---------------------------------------------------------------------------
// Step kernel 2: out[t] = softmax(relu(temp @ fc2_w^T + fc2_b))
// grid 4 x 256 (8 waves); block owns 16 batch rows x full N=1024 so softmax
// fuses. 8 chunks of 128 cols; wave w owns cols [c*128 + w*16, +16).
// ---------------------------------------------------------------------------
__global__ __launch_bounds__(256) void rnn_step2(
    const float* __restrict__ temp, const float* __restrict__ fc2_w,
    const float* __restrict__ fc2_b, float* __restrict__ outp)
{
  __shared__ float As2[2][16][34];
  __shared__ float Bs2[2][128][34];
  __shared__ float redmax[16][8];
  __shared__ float redsum[16][8];
  const int tid  = threadIdx.x;
  const int lane = tid & 31;
  const int wave = tid >> 5;
  const int l16 = lane & 15, hi = lane >> 4;
  const int b0  = blockIdx.x * 16;
  const float* Arow = temp + (size_t)b0 * Hsz;
  const v8f vz = {0.f, 0.f, 0.f, 0.f, 0.f, 0.f, 0.f, 0.f};

  v8f acc[8] = {vz, vz, vz, vz, vz, vz, vz, vz};   // one per 128-col chunk

  stage2(Arow, fc2_w, 0, As2[0], Bs2[0]);
  #pragma unroll
  for (int c = 0; c < 8; ++c) {
    for (int k0 = 0; k0 < Hsz; k0 += 32) {
      const int p = (k0 >> 5) & 1;       // Hsz/32 is even -> parity consistent
      wait_async0();
      __syncthreads();
      if (k0 + 32 < Hsz)
        stage2(Arow, fc2_w + (size_t)(c * 128) * Hsz, k0 + 32,
               As2[p ^ 1], Bs2[p ^ 1]);
      else if (c < 7)
        stage2(Arow, fc2_w + (size_t)((c + 1) * 128) * Hsz, 0,
               As2[p ^ 1], Bs2[p ^ 1]);
      #pragma unroll
      for (int kk = 0; kk < 8; ++kk) {
        const int k = kk * 4 + hi * 2;
        v2f af;
        af.x = As2[p][l16][k];
        af.y = As2[p][l16][k + 1];
        v2f bf;
        bf.x = Bs2[p][wave * 16 + l16][k];
        bf.y = Bs2[p][wave * 16 + l16][k + 1];
        acc[c] = wmma_f32(af, bf, acc[c]);
      }
    }
  }
  __syncthreads();

  // bias + relu, per-thread row maxima (thread's 8 rows fixed by hi)
  float rmax[8];
  #pragma unroll
  for (int r = 0; r < 8; ++r) rmax[r] = -1e30f;
  #pragma unroll
  for (int c = 0; c < 8; ++c) {
    int colb = c * 128 + wave * 16 + l16;
    float bb = fc2_b[colb];
    #pragma unroll
    for (int r = 0; r < 8; ++r) {
      float v = acc[c][r] + bb;
      v = v > 0.f ? v : 0.f;
      acc[c][r] = v;
      rmax[r] = fmaxf(rmax[r], v);
    }
  }
  // reduce across the 16 lanes sharing hi, then across 8 waves via LDS
  #pragma unroll
  for (int mk = 1; mk < 16; mk <<= 1)
    #pragma unroll
    for (int r = 0; r < 8; ++r)
      rmax[r] = fmaxf(rmax[r], __shfl_xor(rmax[r], mk, 32));
  if (l16 == 0)
    #pragma unroll
    for (int r = 0; r < 8; ++r) redmax[hi * 8 + r][wave] = rmax[r];
  __syncthreads();
  float gmax[8];
  #pragma unroll
  for (int r = 0; r < 8; ++r) {
    float mx = redmax[hi * 8 + r][0];
    #pragma unroll
    for (int w = 1; w < 8; ++w) mx = fmaxf(mx, redmax[hi * 8 + r][w]);
    gmax[r] = mx;
  }

  // exp + deterministic fixed-order sum
  float rsum[8];
  #pragma unroll
  for (int r = 0; r < 8; ++r) rsum[r] = 0.f;
  #pragma unroll
  for (int c = 0; c < 8; ++c)
    #pragma unroll
    for (int r = 0; r < 8; ++r) {
      float e = __expf(acc[c][r] - gmax[r]);
      acc[c][r] = e;
      rsum[r] += e;
    }
  #pragma unroll
  for (int mk = 1; mk < 16; mk <<= 1)
    #pragma unroll
    for (int r = 0; r < 8; ++r)
      rsum[r] += __shfl_xor(rsum[r], mk, 32);
  if (l16 == 0)
    #pragma unroll
    for (int r = 0; r < 8; ++r) redsum[hi * 8 + r][wave] = rsum[r];
  __syncthreads();
  #pragma unroll
  for (int r = 0; r < 8; ++r) {
    float sm = 0.f;
    #pragma unroll
    for (int w = 0; w < 8; ++w) sm += redsum[hi * 8 + r][w];
    gmax[r] = 1.f / sm;   // reuse as inverse sum
  }

  #pragma unroll
  for (int c = 0; c < 8; ++c) {
    int colb = c * 128 + wave * 16 + l16;
    #pragma unroll
    for (int r = 0; r < 8; ++r) {
      int row = b0 + hi * 8 + r;
      outp[(size_t)row * Osz + colb] = acc[c][r] * gmax[r];
    }
  }
}

// ---------------------------------------------------------------------------
extern "C" void kernel_launch(void* const* d_in, const int* in_sizes, int n_in,
                              void* d_out, int out_size, void* d_ws, size_t ws_size,
                              hipStream_t stream)
{
  (void)in_sizes; (void)n_in; (void)out_size; (void)ws_size;
  const float* x     = (const float*)d_in[0];   // (B,S,C)
  const float* hid   = (const float*)d_in[1];   // (B,L,H)
  const float* fc1_w = (const float*)d_in[2];   // (H, H+C)
  const float* fc1_b = (const float*)d_in[3];   // (H)
  const float* fc2_w = (const float*)d_in[4];   // (O, H)
  const float* fc2_b = (const float*)d_in[5];   // (O)
  const float* lin_w = (const float*)d_in[6];   // (L-1, H, H)
  const float* lin_b = (const float*)d_in[7];   // (L-1, H)
  float* out = (float*)d_out;                   // (S, B, O)

  float* ws   = (float*)d_ws;
  float* c1   = ws;                             // B*H
  float* hbA  = c1 + Bsz * Hsz;                 // NI*B*H (carry ping)
  float* hbB  = hbA + NI * Bsz * Hsz;           // NI*B*H (carry pong)
  float* temp = hbB + NI * Bsz * Hsz;           // B*H

  rnn_init_c1<<<dim3(Hsz / 64), dim3(128), 0, stream>>>(hid, fc1_w, fc1_b, c1);
  rnn_init_h<<<dim3((NI * Bsz * Hsz + 255) / 256), dim3(256), 0, stream>>>(hid, hbA);

  float* cur = hbA;
  float* nxt = hbB;
  for (int t = 0; t < Ssz; ++t) {
    rnn_step1<<<dim3(Hsz / 64), dim3(128), 0, stream>>>(
        x, c1, cur, fc1_w, lin_w, lin_b, nxt, temp, t);
    rnn_step2<<<dim3(Bsz / 16), dim3(256), 0, stream>>>(
        temp, fc2_w, fc2_b, out + (size_t)t * Bsz * Osz);
    float* sw = cur; cur = nxt; nxt = sw;
  }
}